// CustomCheckMessageGNNLayer_41815801594354
// MI455X (gfx1250) — compile-verified
//
#include <hip/hip_runtime.h>

// CDNA5 / gfx1250. Wave32. f32 WMMA 16x16x4 used for the batched matvec
// (llrs = features @ proj_w + b) fused into the bandwidth-bound copy pass.

typedef float v2f __attribute__((ext_vector_type(2)));
typedef float v4f __attribute__((ext_vector_type(4)));
typedef float v8f __attribute__((ext_vector_type(8)));

#define BIGV 1e30f

// ---------------------------------------------------------------------------
// Pass A: stream features -> out (channel 0 zeroed), compute llrs via WMMA.
// One wave processes a tile of 16 messages (16 x 64 f32 = 4KB) per iteration.
// ---------------------------------------------------------------------------
__global__ __launch_bounds__(256) void llr_copy_kernel(
    const float* __restrict__ feats,   // [B, M, 64]
    const float* __restrict__ w,       // [64]
    const float* __restrict__ bias_p,  // [1]
    float* __restrict__ out,           // [B, M, 64]
    float* __restrict__ llrs,          // [B, M] (workspace)
    int Bn, int M)
{
    const int lane = threadIdx.x & 31;
    const int hi   = (lane >> 4) & 1;          // 0: lanes 0-15, 1: lanes 16-31
    const int waveInBlock   = threadIdx.x >> 5;
    const int wavesPerBlock = blockDim.x >> 5;
    const long long wave   = (long long)blockIdx.x * wavesPerBlock + waveInBlock;
    const long long nWaves = (long long)gridDim.x * wavesPerBlock;
    const int tilesPerB = M >> 4;              // assumes M % 16 == 0 (393216 is)
    const long long T = (long long)Bn * tilesPerB;

    const float bias = bias_p[0];

    // Preload proj_w as B-matrix fragments once per wave. B is 4x16 (KxN);
    // every column identical => VGPR0 = w[4c + 2*hi] (K=0 / K=2),
    // VGPR1 = w[4c + 2*hi + 1] (K=1 / K=3), matching the ISA A/B striping.
    v2f wfrag[16];
    #pragma unroll
    for (int c = 0; c < 16; ++c) {
        wfrag[c].x = w[4 * c + 2 * hi];
        wfrag[c].y = w[4 * c + 2 * hi + 1];
    }

    for (long long t = wave; t < T; t += nWaves) {
        const int b  = (int)(t / tilesPerB);
        const int m0 = (int)(t % tilesPerB) << 4;
        const long long base = ((long long)b * M + m0) * 64;
        const float* src = feats + base;
        float*       dst = out   + base;

        // (1) Coalesced 4KB copy, zeroing channel 0 of each message.
        //     Channel 0 of local message q sits at float4 index q*16.
        const v4f* src4 = (const v4f*)src;
        v4f*       dst4 = (v4f*)dst;
        #pragma unroll
        for (int i = 0; i < 8; ++i) {
            int idx = i * 32 + lane;
            v4f v = src4[idx];
            if ((idx & 15) == 0) v.x = 0.0f;
            dst4[idx] = v;
        }

        // (2) llr matvec: accumulate 16 K-chunks of 4 with f32 WMMA.
        //     A fragment per ISA layout: lane L(<16) holds A[L, 2*hi + {0,1}].
        v8f acc;
        #pragma unroll
        for (int i = 0; i < 8; ++i) acc[i] = bias;   // D = A*B + C carries bias

        const v2f* src2 = (const v2f*)src;
        #pragma unroll
        for (int c = 0; c < 16; ++c) {
            v2f a = src2[(lane & 15) * 32 + c * 2 + hi];  // hits L0 (copy warmed)
            acc = __builtin_amdgcn_wmma_f32_16x16x4_f32(
                /*neg_a=*/false, a, /*neg_b=*/false, wfrag[c],
                /*c_mod=*/(short)0, acc, /*reuse_a=*/false, /*reuse_b=*/false);
        }

        // (3) All B columns identical => lane 0 holds llr[m0+0..7] in acc[0..7],
        //     lane 16 holds llr[m0+8..15].
        if ((lane & 15) == 0) {
            float* lp = llrs + (long long)b * M + m0 + hi * 8;
            v4f lo, hv;
            lo.x = acc[0]; lo.y = acc[1]; lo.z = acc[2]; lo.w = acc[3];
            hv.x = acc[4]; hv.y = acc[5]; hv.z = acc[6]; hv.w = acc[7];
            *(v4f*)(lp)     = lo;
            *(v4f*)(lp + 4) = hv;
        }
    }
}

// ---------------------------------------------------------------------------
// Pass B: min-sum check-node update; one thread per (b, check).
// Scatters alpha * loo_sign * loo_min into out[b, idx, 0] (pre-zeroed).
// ---------------------------------------------------------------------------
__global__ __launch_bounds__(256) void check_kernel(
    const float* __restrict__ llrs,          // [B, M]
    const long long* __restrict__ cidx,      // [C, 8] int64
    const float* __restrict__ alpha_p,       // [1]
    float* __restrict__ out,                 // [B, M, 64]
    int Bn, int M, int C)
{
    long long t = (long long)blockIdx.x * blockDim.x + threadIdx.x;
    long long total = (long long)Bn * C;
    if (t >= total) return;
    const int c = (int)(t % C);
    const int b = (int)(t / C);
    const float alpha = alpha_p[0];

    long long idxs[8];
    float sgn[8], mag[8];
    int   mk[8];
    float min1 = BIGV, min2 = BIGV;
    float tot_sign = 1.0f;
    int   deg = 0;

    #pragma unroll
    for (int j = 0; j < 8; ++j) {
        long long raw = cidx[(long long)c * 8 + j];
        int m = (raw != -1);
        long long id = raw < 0 ? 0 : (raw > (long long)(M - 1) ? (long long)(M - 1) : raw);
        idxs[j] = id;
        mk[j] = m;
        float g = llrs[(long long)b * M + id];
        // jnp.sign semantics: sign(0) == 0
        float s = (g > 0.0f) ? 1.0f : ((g < 0.0f) ? -1.0f : 0.0f);
        sgn[j] = m ? s : 1.0f;
        mag[j] = m ? fabsf(g) : BIGV;
        deg += m;
        tot_sign *= sgn[j];
        float v = mag[j];
        if (v < min1) { min2 = min1; min1 = v; }
        else if (v < min2) { min2 = v; }
    }

    if (deg < 2) return;   // 'write' requires deg >= 2; skip -> channel 0 stays 0

    #pragma unroll
    for (int j = 0; j < 8; ++j) {
        if (!mk[j]) continue;
        float loo_min = (mag[j] == min1) ? min2 : min1;   // ties -> min2, as in ref
        float val = alpha * (tot_sign * sgn[j]) * loo_min;
        out[((long long)b * M + idxs[j]) * 64] = val;
    }
}

// ---------------------------------------------------------------------------
extern "C" void kernel_launch(void* const* d_in, const int* in_sizes, int n_in,
                              void* d_out, int out_size, void* d_ws, size_t ws_size,
                              hipStream_t stream) {
    const float*     feats = (const float*)d_in[0];       // [B, M, 64] f32
    // d_in[1] = message_types : unused by the reference
    const long long* cidx  = (const long long*)d_in[2];   // [C, 8] int64
    const float*     w     = (const float*)d_in[3];       // [64]
    const float*     biasp = (const float*)d_in[4];       // [1]
    const float*     alpha = (const float*)d_in[5];       // [1]
    float*           out   = (float*)d_out;
    float*           llrs  = (float*)d_ws;                // B*M floats scratch

    const int H = 64;
    const int M  = in_sizes[1];                           // message_types is [M]
    const int CD = in_sizes[2];
    const int C  = CD / 8;
    const int Bn = (int)((long long)in_sizes[0] / ((long long)M * H));

    // Pass A: bandwidth-bound stream (copy + zero ch0 + WMMA matvec).
    llr_copy_kernel<<<2048, 256, 0, stream>>>(feats, w, biasp, out, llrs, Bn, M);

    // Pass B: tiny check-node scatter.
    long long totalChecks = (long long)Bn * C;
    int blocks = (int)((totalChecks + 255) / 256);
    check_kernel<<<blocks, 256, 0, stream>>>(llrs, cidx, alpha, out, Bn, M, C);
}